// CompressionAttention_8358006358682
// MI455X (gfx1250) — compile-verified
//
#include <hip/hip_runtime.h>
#include <hip/hip_bf16.h>
#include <math.h>

// ---------------------------------------------------------------------------
// CompressionAttention for MI455X (gfx1250, wave32, WMMA)
// S=1024 B=1 D=1024 H=16 C=64 DH=64
// ---------------------------------------------------------------------------

typedef __attribute__((ext_vector_type(2))) float v2f;
typedef __attribute__((ext_vector_type(8))) float v8f;

#define S_LEN 1024
#define D_MOD 1024
#define H_NUM 16
#define C_LEN 64
#define DH_L  64
#define NCH   32      // number of s-chunks for the scan
#define CH_L  32      // chunk length (NCH*CH_L == S_LEN)
#define VLEN  132     // padded per-c vector in chunk buffer: 128 kv + 1 norm + pad

#define WMMA_F32(a, b, c) \
    __builtin_amdgcn_wmma_f32_16x16x4_f32(false, (a), false, (b), (short)0, (c), false, false)

// ---------------------------------------------------------------------------
// Generic f32 WMMA GEMM:  C[m][n] = sum_k A[m*lda + k] * Bt[n*ldb + k] (+bias[n])
// V_WMMA_F32_16X16X4_F32; one wave computes a 32(M) x 64(N) tile with a
// one-step software pipeline (fragments for k+4 are in flight while the
// 8 WMMAs for k execute on the matrix pipe).
// blockDim.x = 128 (4 waves, 2x2 wave tiling) -> block covers 64 M x 128 N.
// grid: x = N/128, y = M/64, z = batch (per-z element offsets).
// ---------------------------------------------------------------------------
__global__ void gemm_xwT_wmma(const float* __restrict__ A, int lda, long long aOffZ,
                              const float* __restrict__ Bt, int ldb, long long bOffZ,
                              const float* __restrict__ bias,
                              float* __restrict__ C, int ldc, long long cOffZ,
                              int Kdim)
{
    const int lane = threadIdx.x & 31;
    const int wave = threadIdx.x >> 5;
    const long long z = blockIdx.z;

    A  += z * aOffZ;
    Bt += z * bOffZ;
    C  += z * cOffZ;

    const int mw = wave & 1, nw = wave >> 1;
    const int m0 = blockIdx.y * 64 + mw * 32;
    const int n0 = blockIdx.x * 128 + nw * 64;

    // A fragment: lane l holds row M = l&15, K = k + 2*(l>>4) + {0,1}
    const int am = lane & 15;
    const int kh = (lane >> 4) * 2;
    // B fragment: lane l holds B[k + 2*(l>>4) + v][n + (l&15)] = Bt[(n+cn)*ldb + ...]
    const int cn = lane & 15;

    const float* arow0 = A + (long long)(m0 +  0 + am) * lda + kh;
    const float* arow1 = A + (long long)(m0 + 16 + am) * lda + kh;
    const float* br0   = Bt + (long long)(n0 +  0 + cn) * ldb + kh;
    const float* br1   = Bt + (long long)(n0 + 16 + cn) * ldb + kh;
    const float* br2   = Bt + (long long)(n0 + 32 + cn) * ldb + kh;
    const float* br3   = Bt + (long long)(n0 + 48 + cn) * ldb + kh;

    v8f acc[2][4];
#pragma unroll
    for (int i = 0; i < 2; ++i)
#pragma unroll
        for (int j = 0; j < 4; ++j) acc[i][j] = (v8f){};

    // prologue: fragments for k = 0
    v2f a0 = *(const v2f*)(arow0);
    v2f a1 = *(const v2f*)(arow1);
    v2f b0 = *(const v2f*)(br0);
    v2f b1 = *(const v2f*)(br1);
    v2f b2 = *(const v2f*)(br2);
    v2f b3 = *(const v2f*)(br3);

    for (int k = 0; k < Kdim - 4; k += 4) {
        // issue next k-step's loads before consuming current fragments
        const int kn = k + 4;
        v2f a0n = *(const v2f*)(arow0 + kn);
        v2f a1n = *(const v2f*)(arow1 + kn);
        v2f b0n = *(const v2f*)(br0 + kn);
        v2f b1n = *(const v2f*)(br1 + kn);
        v2f b2n = *(const v2f*)(br2 + kn);
        v2f b3n = *(const v2f*)(br3 + kn);

        acc[0][0] = WMMA_F32(a0, b0, acc[0][0]);
        acc[0][1] = WMMA_F32(a0, b1, acc[0][1]);
        acc[0][2] = WMMA_F32(a0, b2, acc[0][2]);
        acc[0][3] = WMMA_F32(a0, b3, acc[0][3]);
        acc[1][0] = WMMA_F32(a1, b0, acc[1][0]);
        acc[1][1] = WMMA_F32(a1, b1, acc[1][1]);
        acc[1][2] = WMMA_F32(a1, b2, acc[1][2]);
        acc[1][3] = WMMA_F32(a1, b3, acc[1][3]);

        a0 = a0n; a1 = a1n;
        b0 = b0n; b1 = b1n; b2 = b2n; b3 = b3n;
    }
    // epilogue: last k-step
    acc[0][0] = WMMA_F32(a0, b0, acc[0][0]);
    acc[0][1] = WMMA_F32(a0, b1, acc[0][1]);
    acc[0][2] = WMMA_F32(a0, b2, acc[0][2]);
    acc[0][3] = WMMA_F32(a0, b3, acc[0][3]);
    acc[1][0] = WMMA_F32(a1, b0, acc[1][0]);
    acc[1][1] = WMMA_F32(a1, b1, acc[1][1]);
    acc[1][2] = WMMA_F32(a1, b2, acc[1][2]);
    acc[1][3] = WMMA_F32(a1, b3, acc[1][3]);

    // C/D layout: VGPR r -> row mtile + r + 8*(lane>>4), column n-base + (lane&15)
#pragma unroll
    for (int mi = 0; mi < 2; ++mi) {
        const int mrow = m0 + mi * 16 + (lane >> 4) * 8;
#pragma unroll
        for (int j = 0; j < 4; ++j) {
            const int col = n0 + j * 16 + cn;
            const float bv = bias ? bias[col] : 0.0f;
#pragma unroll
            for (int r = 0; r < 8; ++r) {
                C[(long long)(mrow + r) * ldc + col] = acc[mi][j][r] + bv;
            }
        }
    }
}

// ---------------------------------------------------------------------------
// Row max over s for cw_raw: rows = H*C = 1024, each of length S.
// ---------------------------------------------------------------------------
__global__ void rowmax_kernel(const float* __restrict__ cw, float* __restrict__ mx)
{
    __shared__ float red[256];
    const int row = blockIdx.x;
    const float* p = cw + (long long)row * S_LEN;
    float m = -INFINITY;
    for (int i = threadIdx.x; i < S_LEN; i += 256) m = fmaxf(m, p[i]);
    red[threadIdx.x] = m;
    __syncthreads();
    for (int s = 128; s > 0; s >>= 1) {
        if (threadIdx.x < s) red[threadIdx.x] = fmaxf(red[threadIdx.x], red[threadIdx.x + s]);
        __syncthreads();
    }
    if (threadIdx.x == 0) mx[row] = red[0];
}

// ---------------------------------------------------------------------------
// Phase A of the chunked scan: per (head, chunk) compute
//   chunk[c][d]   = sum_{t in chunk} w[c][t] * kv[t][d]   (d in 0..127: k|v)
//   chunk[c][128] = sum_{t in chunk} w[c][t]
// grid (NCH, H), blockDim 256.
// ---------------------------------------------------------------------------
__global__ void chunk_sum_kernel(const float* __restrict__ K, const float* __restrict__ V,
                                 const float* __restrict__ cw, const float* __restrict__ mx,
                                 float* __restrict__ chunkbuf)
{
    const int ch = blockIdx.x, h = blockIdx.y;
    const int s0 = ch * CH_L;

    __shared__ float kv[CH_L][128];
    __shared__ float w[C_LEN][CH_L + 1];

    for (int i = threadIdx.x; i < CH_L * 128; i += 256) {
        const int t = i >> 7, d = i & 127;
        const float* src = (d < 64) ? K : V;
        kv[t][d] = src[(long long)(s0 + t) * D_MOD + h * DH_L + (d & 63)];
    }
    for (int i = threadIdx.x; i < C_LEN * CH_L; i += 256) {
        const int c = i >> 5, t = i & 31;
        const int row = h * C_LEN + c;
        w[c][t] = __expf(cw[(long long)row * S_LEN + s0 + t] - mx[row]);
    }
    __syncthreads();

    float* out = chunkbuf + (long long)((h * NCH + ch) * C_LEN) * VLEN;
    for (int i = threadIdx.x; i < C_LEN * 128; i += 256) {
        const int c = i >> 7, d = i & 127;
        float sum = 0.0f;
#pragma unroll
        for (int t = 0; t < CH_L; ++t) sum += w[c][t] * kv[t][d];
        out[c * VLEN + d] = sum;
    }
    for (int c = threadIdx.x; c < C_LEN; c += 256) {
        float sum = 0.0f;
#pragma unroll
        for (int t = 0; t < CH_L; ++t) sum += w[c][t];
        out[c * VLEN + 128] = sum;
    }
}

// ---------------------------------------------------------------------------
// Phase B: in-place exclusive scan of chunk sums across the NCH chunks,
// independently per (head, c, d). grid (C, H), blockDim 160 (129 active).
// ---------------------------------------------------------------------------
__global__ void chunk_scan_kernel(float* __restrict__ chunkbuf)
{
    const int c = blockIdx.x, h = blockIdx.y;
    const int d = threadIdx.x;
    if (d >= 129) return;
    float pref = 0.0f;
    for (int ch = 0; ch < NCH; ++ch) {
        float* p = chunkbuf + (long long)((h * NCH + ch) * C_LEN + c) * VLEN + d;
        const float tmp = *p;
        *p = pref;
        pref += tmp;
    }
}

// ---------------------------------------------------------------------------
// Phase C: per (head, chunk) local sequential scan seeded with the exclusive
// prefix; at each position t compute logits vs the running compressed keys,
// softmax over C, and the output head vector. grid (NCH, H), blockDim 256.
// ---------------------------------------------------------------------------
__global__ void attend_kernel(const float* __restrict__ Q, const float* __restrict__ K,
                              const float* __restrict__ V, const float* __restrict__ cw,
                              const float* __restrict__ mx, const float* __restrict__ beta,
                              const float* __restrict__ chunkbuf, float* __restrict__ O)
{
    const int ch = blockIdx.x, h = blockIdx.y;
    const int s0 = ch * CH_L;
    const int tid = threadIdx.x;

    __shared__ float acc[C_LEN][130];   // running sums: [0..127]=w*kv, [128]=norm
    __shared__ float kvs[CH_L][128];
    __shared__ float qs[CH_L][DH_L];
    __shared__ float wcur[C_LEN];
    __shared__ float lred[C_LEN];
    __shared__ float red[C_LEN];
    __shared__ float swv[C_LEN];
    __shared__ float part[4][DH_L];

    const float scale = 0.125f * __expf(-beta[h]);   // 1/(sqrt(64)*exp(beta_h))

    // seed running state from exclusive chunk prefix
    const float* pre = chunkbuf + (long long)((h * NCH + ch) * C_LEN) * VLEN;
    for (int i = tid; i < C_LEN * 129; i += 256) {
        const int c = i / 129, d = i % 129;
        acc[c][d] = pre[c * VLEN + d];
    }
    for (int i = tid; i < CH_L * 128; i += 256) {
        const int t = i >> 7, d = i & 127;
        const float* src = (d < 64) ? K : V;
        kvs[t][d] = src[(long long)(s0 + t) * D_MOD + h * DH_L + (d & 63)];
    }
    for (int i = tid; i < CH_L * DH_L; i += 256) {
        const int t = i >> 6, d = i & 63;
        qs[t][d] = Q[(long long)(s0 + t) * D_MOD + h * DH_L + d];
    }
    __syncthreads();

    for (int t = 0; t < CH_L; ++t) {
        // cw weight at this position, per compressed slot c
        if (tid < C_LEN) {
            const int row = h * C_LEN + tid;
            wcur[tid] = __expf(cw[(long long)row * S_LEN + s0 + t] - mx[row]);
        }
        __syncthreads();

        // running weighted sums (inclusive at position t)
        for (int i = tid; i < C_LEN * 128; i += 256) {
            const int c = i >> 7, d = i & 127;
            acc[c][d] += wcur[c] * kvs[t][d];
        }
        if (tid < C_LEN) acc[tid][128] += wcur[tid];
        __syncthreads();

        // logits: q[t] . (accK[c]/norm[c]) * scale
        if (tid < C_LEN) {
            const int c = tid;
            float dot = 0.0f;
#pragma unroll
            for (int d = 0; d < DH_L; ++d) dot += qs[t][d] * acc[c][d];
            const float l = dot * scale / acc[c][128];
            lred[c] = l;
            red[c]  = l;
        }
        __syncthreads();
        // max over 64
        for (int sdiv = 32; sdiv >= 1; sdiv >>= 1) {
            if (tid < sdiv) red[tid] = fmaxf(red[tid], red[tid + sdiv]);
            __syncthreads();
        }
        const float smax = red[0];
        __syncthreads();
        if (tid < C_LEN) {
            const float e = __expf(lred[tid] - smax);
            swv[tid] = e;
            red[tid] = e;
        }
        __syncthreads();
        for (int sdiv = 32; sdiv >= 1; sdiv >>= 1) {
            if (tid < sdiv) red[tid] += red[tid + sdiv];
            __syncthreads();
        }
        const float ssum = red[0];
        __syncthreads();
        if (tid < C_LEN) {
            // fold softmax normalization and 1/norm[c] (v_c = accV/norm) together
            swv[tid] = swv[tid] / (ssum * acc[tid][128]);
        }
        __syncthreads();

        // o[t][d] = sum_c swv[c] * accV[c][d]
        {
            const int d = tid & 63, grp = tid >> 6;
            float p = 0.0f;
#pragma unroll
            for (int cc = 0; cc < 16; ++cc) p += swv[grp * 16 + cc] * acc[grp * 16 + cc][64 + d];
            part[grp][d] = p;
        }
        __syncthreads();
        if (tid < DH_L) {
            O[(long long)(s0 + t) * D_MOD + h * DH_L + tid] =
                part[0][tid] + part[1][tid] + part[2][tid] + part[3][tid];
        }
        __syncthreads();
    }
}

// ---------------------------------------------------------------------------
// Launch pipeline (graph-capture safe: only kernel launches on `stream`).
// ---------------------------------------------------------------------------
extern "C" void kernel_launch(void* const* d_in, const int* in_sizes, int n_in,
                              void* d_out, int out_size, void* d_ws, size_t ws_size,
                              hipStream_t stream)
{
    (void)in_sizes; (void)n_in; (void)out_size; (void)ws_size;

    const float* x    = (const float*)d_in[0];   // (S,1,D) == (1024,1024)
    const float* q_c  = (const float*)d_in[1];   // (C,D)
    const float* beta = (const float*)d_in[2];   // (H,)
    const float* Wq   = (const float*)d_in[3];
    const float* bq   = (const float*)d_in[4];
    const float* Wk   = (const float*)d_in[5];
    const float* bk   = (const float*)d_in[6];
    const float* Wv   = (const float*)d_in[7];
    const float* bv   = (const float*)d_in[8];
    const float* Wo   = (const float*)d_in[9];
    const float* bo   = (const float*)d_in[10];
    float* out = (float*)d_out;

    float* ws = (float*)d_ws;
    const size_t MAT = (size_t)S_LEN * D_MOD;              // 1M floats
    float* Qm    = ws;                                      // (S, D)
    float* Km    = ws + MAT;                                // (S, D)
    float* Vm    = ws + 2 * MAT;                            // (S, D)
    float* CWm   = ws + 3 * MAT;                            // (H*C, S)
    float* MXm   = ws + 4 * MAT;                            // (H*C)
    float* Om    = ws + 4 * MAT + 4096;                     // (S, D)
    float* CHUNK = ws + 5 * MAT + 4096;                     // H*NCH*C*VLEN

    const dim3 gblk(128);

    // Q/K/V projections: (1024x1024) @ (1024x1024)^T + bias
    // block covers 64 M x 128 N -> grid (N/128, M/64)
    gemm_xwT_wmma<<<dim3(8, 16, 1), gblk, 0, stream>>>(x, D_MOD, 0, Wq, D_MOD, 0, bq, Qm, D_MOD, 0, D_MOD);
    gemm_xwT_wmma<<<dim3(8, 16, 1), gblk, 0, stream>>>(x, D_MOD, 0, Wk, D_MOD, 0, bk, Km, D_MOD, 0, D_MOD);
    gemm_xwT_wmma<<<dim3(8, 16, 1), gblk, 0, stream>>>(x, D_MOD, 0, Wv, D_MOD, 0, bv, Vm, D_MOD, 0, D_MOD);

    // cw_raw[h*C + c][s] = qc[h,c,:] . k[s, h*64:...]; per-head M=64,N=1024,K=64
    gemm_xwT_wmma<<<dim3(8, 1, H_NUM), gblk, 0, stream>>>(q_c, D_MOD, DH_L,
                                                         Km, D_MOD, DH_L,
                                                         nullptr,
                                                         CWm, S_LEN, (long long)C_LEN * S_LEN,
                                                         DH_L);

    rowmax_kernel<<<dim3(H_NUM * C_LEN), 256, 0, stream>>>(CWm, MXm);

    chunk_sum_kernel<<<dim3(NCH, H_NUM), 256, 0, stream>>>(Km, Vm, CWm, MXm, CHUNK);
    chunk_scan_kernel<<<dim3(C_LEN, H_NUM), 160, 0, stream>>>(CHUNK);
    attend_kernel<<<dim3(NCH, H_NUM), 256, 0, stream>>>(Qm, Km, Vm, CWm, MXm, beta, CHUNK, Om);

    // output projection
    gemm_xwT_wmma<<<dim3(8, 16, 1), gblk, 0, stream>>>(Om, D_MOD, 0, Wo, D_MOD, 0, bo, out, D_MOD, 0, D_MOD);
}